// SimpleGNN_60713657696826
// MI455X (gfx1250) — compile-verified
//
#include <hip/hip_runtime.h>

#define D 128

typedef __attribute__((ext_vector_type(2))) float v2f;
typedef __attribute__((ext_vector_type(8))) float v8f;

// ---------------- small utility kernels ----------------

__global__ __launch_bounds__(256) void k_init_deg(float* deg, int n) {
  int i = blockIdx.x * 256 + threadIdx.x;
  if (i < n) deg[i] = 1.0f;  // self-loop
}

__global__ __launch_bounds__(256) void k_count_deg(const int* __restrict__ dst,
                                                   float* deg, int nE) {
  int e = blockIdx.x * 256 + threadIdx.x;
  if (e < nE) atomicAdd(&deg[dst[e]], 1.0f);
}

__global__ __launch_bounds__(256) void k_rsqrt(float* deg, int n) {
  int i = blockIdx.x * 256 + threadIdx.x;
  if (i < n) deg[i] = rsqrtf(deg[i]);  // in-place: deg -> dinv
}

__global__ __launch_bounds__(256) void k_zero(float* p, int n) {
  int i = blockIdx.x * 256 + threadIdx.x;
  if (i < n) p[i] = 0.0f;
}

// ---------------- fp32 WMMA GEMM: C[M x 128] = A[M x 128] * W[128 x 128] ----
// One wave32 per 16x16 output tile; 8 waves per block cover all 128 columns
// of a 16-row strip. W (64KB) staged in LDS and shared by the 8 waves.

__global__ __launch_bounds__(256) void k_gemm128_wmma(
    const float* __restrict__ A, const float* __restrict__ W,
    float* __restrict__ Cmat, int M) {
  __shared__ float sW[D * D];  // 64 KB
  {
    const float4* Wv = (const float4*)W;
    float4* sWv = (float4*)sW;
#pragma unroll
    for (int i = 0; i < (D * D / 4) / 256; ++i)
      sWv[threadIdx.x + i * 256] = Wv[threadIdx.x + i * 256];
  }
  __syncthreads();

  const int lane = threadIdx.x & 31;
  const int wave = threadIdx.x >> 5;
  const int row0 = blockIdx.x * 16;
  const int col0 = wave * 16;

  // A fragment (16x4 f32): lanes 0-15 -> K={0,1}, lanes 16-31 -> K={2,3}
  const int mi = lane & 15;
  const int kh = lane >> 4;
  int arow = row0 + mi;
  if (arow >= M) arow = M - 1;  // clamp (M is a multiple of 16 here anyway)
  const float* Arow = A + (size_t)arow * D;

  v8f acc = {};
#pragma unroll
  for (int k = 0; k < D; k += 4) {
    const int ka = k + 2 * kh;                   // even -> 8B aligned
    v2f a = *(const v2f*)(Arow + ka);            // A[m][ka], A[m][ka+1]
    v2f b;
    b.x = sW[ka * D + col0 + mi];                // B[ka][n]
    b.y = sW[(ka + 1) * D + col0 + mi];          // B[ka+1][n]
    acc = __builtin_amdgcn_wmma_f32_16x16x4_f32(
        /*neg_a=*/false, a, /*neg_b=*/false, b,
        /*c_mod=*/(short)0, acc, /*reuse_a=*/false, /*reuse_b=*/false);
  }

  // C/D layout: VGPR r -> M=r (lanes 0-15), M=r+8 (lanes 16-31); N = lane&15
  const int mh = kh;
#pragma unroll
  for (int r = 0; r < 8; ++r) {
    int row = row0 + r + 8 * mh;
    if (row < M) Cmat[(size_t)row * D + col0 + mi] = acc[r];
  }
}

// ---------------- edge scatter, 128-wide features ----------------
// One wave per edge iteration: 32 lanes x float4 = 128 floats/edge.

__global__ __launch_bounds__(256) void k_scatter128(
    const float* __restrict__ h, const float* __restrict__ dinv,
    const int* __restrict__ src, const int* __restrict__ dst,
    float* __restrict__ agg, int nE) {
  const int lane = threadIdx.x & 31;
  const int wid = blockIdx.x * (blockDim.x >> 5) + (threadIdx.x >> 5);
  const int nw = gridDim.x * (blockDim.x >> 5);
  for (int e = wid; e < nE; e += nw) {
    const int s = src[e];
    const int d = dst[e];
    const float w = dinv[s] * dinv[d];
    const float4 hv = ((const float4*)(h + (size_t)s * D))[lane];
    float* out = agg + (size_t)d * D + lane * 4;
    atomicAdd(out + 0, hv.x * w);
    atomicAdd(out + 1, hv.y * w);
    atomicAdd(out + 2, hv.z * w);
    atomicAdd(out + 3, hv.w * w);
  }
}

// ---------------- out = [relu](agg + h*dinv^2 + b) ----------------

__global__ __launch_bounds__(256) void k_fuse(
    const float* __restrict__ agg, const float* __restrict__ h,
    const float* __restrict__ dinv, const float* __restrict__ bias,
    float* __restrict__ out, int n, int relu) {
  int idx = blockIdx.x * 256 + threadIdx.x;  // node*32 + f4
  if (idx >= n * (D / 4)) return;
  int node = idx >> 5;
  int f4 = idx & 31;
  float s = dinv[node];
  s *= s;
  float4 a = ((const float4*)agg)[idx];
  float4 hv = ((const float4*)h)[idx];
  float4 b = ((const float4*)bias)[f4];
  float4 o;
  o.x = fmaf(hv.x, s, a.x) + b.x;
  o.y = fmaf(hv.y, s, a.y) + b.y;
  o.z = fmaf(hv.z, s, a.z) + b.z;
  o.w = fmaf(hv.w, s, a.w) + b.w;
  if (relu) {
    o.x = fmaxf(o.x, 0.0f);
    o.y = fmaxf(o.y, 0.0f);
    o.z = fmaxf(o.z, 0.0f);
    o.w = fmaxf(o.w, 0.0f);
  }
  ((float4*)out)[idx] = o;
}

// ---------------- layer 3: H3[n x 2] = X[n x 128] * W3[128 x 2] ----------

__global__ __launch_bounds__(256) void k_gemm_small(
    const float* __restrict__ X, const float* __restrict__ W3,
    float* __restrict__ H3, int n) {
  int i = blockIdx.x * 256 + threadIdx.x;
  if (i >= n) return;
  const float4* xr = (const float4*)(X + (size_t)i * D);
  float s0 = 0.0f, s1 = 0.0f;
#pragma unroll
  for (int k4 = 0; k4 < D / 4; ++k4) {
    float4 xv = xr[k4];
    float4 w01 = ((const float4*)W3)[k4 * 2 + 0];  // rows 4k,4k+1
    float4 w23 = ((const float4*)W3)[k4 * 2 + 1];  // rows 4k+2,4k+3
    s0 = fmaf(xv.x, w01.x, s0); s1 = fmaf(xv.x, w01.y, s1);
    s0 = fmaf(xv.y, w01.z, s0); s1 = fmaf(xv.y, w01.w, s1);
    s0 = fmaf(xv.z, w23.x, s0); s1 = fmaf(xv.z, w23.y, s1);
    s0 = fmaf(xv.w, w23.z, s0); s1 = fmaf(xv.w, w23.w, s1);
  }
  H3[(size_t)i * 2 + 0] = s0;
  H3[(size_t)i * 2 + 1] = s1;
}

__global__ __launch_bounds__(256) void k_scatter2(
    const float* __restrict__ h3, const float* __restrict__ dinv,
    const int* __restrict__ src, const int* __restrict__ dst,
    float* __restrict__ agg3, int nE) {
  int e = blockIdx.x * 256 + threadIdx.x;
  if (e >= nE) return;
  int s = src[e], d = dst[e];
  float w = dinv[s] * dinv[d];
  atomicAdd(&agg3[(size_t)d * 2 + 0], h3[(size_t)s * 2 + 0] * w);
  atomicAdd(&agg3[(size_t)d * 2 + 1], h3[(size_t)s * 2 + 1] * w);
}

// ---------------- segment mean pool ----------------

__global__ __launch_bounds__(256) void k_pool(
    const float* __restrict__ agg3, const float* __restrict__ h3,
    const float* __restrict__ dinv, const float* __restrict__ b3,
    const int* __restrict__ batch, float* __restrict__ sums,
    float* __restrict__ cnt, int n) {
  int i = blockIdx.x * 256 + threadIdx.x;
  if (i >= n) return;
  float s = dinv[i];
  s *= s;
  float o0 = fmaf(h3[(size_t)i * 2 + 0], s, agg3[(size_t)i * 2 + 0]) + b3[0];
  float o1 = fmaf(h3[(size_t)i * 2 + 1], s, agg3[(size_t)i * 2 + 1]) + b3[1];
  int g = batch[i];
  atomicAdd(&sums[g * 2 + 0], o0);
  atomicAdd(&sums[g * 2 + 1], o1);
  atomicAdd(&cnt[g], 1.0f);
}

__global__ __launch_bounds__(64) void k_finalize(const float* __restrict__ sums,
                                                 const float* __restrict__ cnt,
                                                 float* __restrict__ out, int nG) {
  int g = threadIdx.x;
  if (g >= nG) return;
  float c = fmaxf(cnt[g], 1.0f);
  out[g * 2 + 0] = sums[g * 2 + 0] / c;
  out[g * 2 + 1] = sums[g * 2 + 1] / c;
}

// ---------------- launcher ----------------

extern "C" void kernel_launch(void* const* d_in, const int* in_sizes, int n_in,
                              void* d_out, int out_size, void* d_ws, size_t ws_size,
                              hipStream_t stream) {
  const float* x = (const float*)d_in[0];
  const int* ei = (const int*)d_in[1];
  const int* batch = (const int*)d_in[2];
  const float* W1 = (const float*)d_in[3];
  const float* b1 = (const float*)d_in[4];
  const float* W2 = (const float*)d_in[5];
  const float* b2 = (const float*)d_in[6];
  const float* W3 = (const float*)d_in[7];
  const float* b3 = (const float*)d_in[8];
  (void)n_in; (void)ws_size;

  const int n = in_sizes[0] / D;       // 100000
  const int nE = in_sizes[1] / 2;      // 1600000
  const int nG = out_size / 2;         // 64
  const int* src = ei;
  const int* dst = ei + nE;

  // workspace layout (256B aligned slots)
  char* ws = (char*)d_ws;
  size_t off = 0;
  auto take = [&](size_t bytes) -> char* {
    char* p = ws + off;
    off = (off + bytes + 255) & ~(size_t)255;
    return p;
  };
  float* dinv = (float*)take((size_t)n * 4);
  float* bufA = (float*)take((size_t)n * D * 4);
  float* bufB = (float*)take((size_t)n * D * 4);
  float* h3 = (float*)take((size_t)n * 2 * 4);
  float* agg3 = (float*)take((size_t)n * 2 * 4);
  float* sums = (float*)take((size_t)nG * 3 * 4);
  float* cnt = sums + nG * 2;

  dim3 blk(256);
  const int gN = (n + 255) / 256;
  const int gE = (nE + 255) / 256;
  const int gNF = (n * (D / 4) + 255) / 256;
  const int gND = (n * D + 255) / 256;
  const int gTile = (n + 15) / 16;

  // degrees -> dinv
  k_init_deg<<<gN, blk, 0, stream>>>(dinv, n);
  k_count_deg<<<gE, blk, 0, stream>>>(dst, dinv, nE);
  k_rsqrt<<<gN, blk, 0, stream>>>(dinv, n);

  // layer 1: bufA = x@W1 ; bufB = scatter ; bufA = relu(fuse)
  k_gemm128_wmma<<<gTile, blk, 0, stream>>>(x, W1, bufA, n);
  k_zero<<<gND, blk, 0, stream>>>(bufB, n * D);
  k_scatter128<<<4096, blk, 0, stream>>>(bufA, dinv, src, dst, bufB, nE);
  k_fuse<<<gNF, blk, 0, stream>>>(bufB, bufA, dinv, b1, bufA, n, 1);

  // layer 2: bufB = bufA@W2 ; bufA = scatter ; bufB = relu(fuse)
  k_gemm128_wmma<<<gTile, blk, 0, stream>>>(bufA, W2, bufB, n);
  k_zero<<<gND, blk, 0, stream>>>(bufA, n * D);
  k_scatter128<<<4096, blk, 0, stream>>>(bufB, dinv, src, dst, bufA, nE);
  k_fuse<<<gNF, blk, 0, stream>>>(bufA, bufB, dinv, b2, bufB, n, 1);

  // layer 3 (D_OUT = 2) + mean pool
  k_gemm_small<<<gN, blk, 0, stream>>>(bufB, W3, h3, n);
  k_zero<<<(2 * n + 255) / 256, blk, 0, stream>>>(agg3, 2 * n);
  k_scatter2<<<gE, blk, 0, stream>>>(h3, dinv, src, dst, agg3, nE);
  k_zero<<<1, blk, 0, stream>>>(sums, nG * 3);
  k_pool<<<gN, blk, 0, stream>>>(agg3, h3, dinv, b3, batch, sums, cnt, n);
  k_finalize<<<1, dim3(64), 0, stream>>>(sums, cnt, (float*)d_out, nG);
}